// PointSeg_44143673869125
// MI455X (gfx1250) — compile-verified
//
#include <hip/hip_runtime.h>
#include <math.h>

// ---------------------------------------------------------------------------
// CDNA5 (gfx1250) PointNet++ part segmentation.
// All dense layers run through V_WMMA_F32_16X16X4_F32 (fp32 WMMA, wave32).
// ---------------------------------------------------------------------------

typedef float v2f __attribute__((ext_vector_type(2)));
typedef float v8f __attribute__((ext_vector_type(8)));

#define NW 8  // waves per 256-thread workgroup (wave32)

struct Mlp3 {
  const float* W[4];   // (cout, cin) row-major
  const float* S[4];   // folded-BN scale (cout)
  const float* Bv[4];  // folded-BN shift (cout)
  int cin[4];
  int cout[4];
  int nl;
};

enum { OUT_LDS = 0, OUT_GLOBAL = 1, OUT_GMAX = 2 };

// One fused GEMM layer: Xout = relu(Xin * W^T * s + b).
// Xin is an LDS activation buffer [rows x ldin] (zero-padded to 4-col mult);
// W streamed from L2 (weights are tiny and L2-resident).
// WMMA fragment mapping (wave32):
//   A 16x4 f32 : lane 0-15 -> M=lane (K pair 0,1), lane 16-31 -> M=lane-16 (K pair 2,3)
//   B 4x16 f32 : lane gives N, half-wave gives K pair (B[k][n] = W[n][k])
//   C/D 16x16  : vgpr v, lane l -> row v + 8*(l>>4), col l&15
template <int MODE>
__device__ __forceinline__ void gemm_layer(
    const float* Xin, int ldin,
    float* Xout, int ldout,                    // MODE==OUT_LDS
    float* gout, int gld, int grow0,           // MODE==OUT_GLOBAL
    int* gmax, int mld, int mrow0,             // MODE==OUT_GMAX (maxpool over k=32)
    int rows, int Cin, int Cout,
    const float* __restrict__ W, const float* __restrict__ sv,
    const float* __restrict__ bv)
{
  const int tid  = threadIdx.x;
  const int wv   = tid >> 5;
  const int lane = tid & 31;
  const int hh   = lane >> 4;
  const int l15  = lane & 15;
  const int RT   = rows >> 4;
  const int CT   = Cout >> 4;
  const int kmain = Cin & ~3;                  // guard-free K range
  for (int t = wv; t < RT * CT; t += NW) {
    const int rt   = t % RT;
    const int ct   = t / RT;
    const int row0 = rt << 4;
    const int n0   = ct << 4;
    v8f acc = {0.f, 0.f, 0.f, 0.f, 0.f, 0.f, 0.f, 0.f};
    const float* xr = Xin + (size_t)(row0 + l15) * ldin;
    const float* wr = W + (size_t)(n0 + l15) * Cin;
    __builtin_prefetch(wr, 0, 0);
    const float sc = sv[n0 + l15];
    const float bc = bv[n0 + l15];
    // main loop: unconditional 8B LDS load (ldin even, ka even -> aligned)
    // + two adjacent global loads; no EXEC games, WMMAs overlap via unroll.
#pragma unroll 4
    for (int k0 = 0; k0 < kmain; k0 += 4) {
      const int ka = k0 + hh * 2;
      v2f av = *(const v2f*)(xr + ka);
      v2f bf;
      bf.x = wr[ka];
      bf.y = wr[ka + 1];
      acc = __builtin_amdgcn_wmma_f32_16x16x4_f32(false, av, false, bf,
                                                  (short)0, acc, false, false);
    }
    if (kmain != Cin) {  // branchless tail (Cin % 4 == 3 layers only), once per tile
      const int ka = kmain + hh * 2;
      v2f av = *(const v2f*)(xr + ka);           // LDS zero-padded: safe
      int ia0 = ka;     if (ia0 >= Cin) ia0 = 0; // clamp addr, mask value
      int ia1 = ka + 1; if (ia1 >= Cin) ia1 = 0;
      v2f bf;
      bf.x = wr[ia0]; if (ka >= Cin)     bf.x = 0.f;
      bf.y = wr[ia1]; if (ka + 1 >= Cin) bf.y = 0.f;
      acc = __builtin_amdgcn_wmma_f32_16x16x4_f32(false, av, false, bf,
                                                  (short)0, acc, false, false);
    }
#pragma unroll
    for (int v = 0; v < 8; ++v) {
      const float val = fmaxf(acc[v] * sc + bc, 0.f);
      const int row = row0 + v + hh * 8;
      const int col = n0 + l15;
      if (MODE == OUT_LDS) {
        Xout[(size_t)row * ldout + col] = val;
      } else if (MODE == OUT_GLOBAL) {
        gout[(size_t)(grow0 + row) * gld + col] = val;
      } else {
        // relu output >= 0 -> float bits are order-preserving ints
        atomicMax(gmax + (size_t)((mrow0 + row) >> 5) * mld + col,
                  __float_as_int(val));
      }
    }
  }
}

// ---------------------------------------------------------------------------
// Farthest point sampling: one workgroup per batch, dist kept in registers.
// Serial bottleneck of the net (1024+256+64+16 dependent argmax iterations),
// so the per-iteration reduction uses wave shuffles: 2 barriers/iter total.
// ---------------------------------------------------------------------------
template <int TPB, int PPT>
__global__ __launch_bounds__(TPB) void fps_kernel(const float* __restrict__ xyz,
                                                  int N, int npoint,
                                                  int* __restrict__ out_idx)
{
  constexpr int NWV = TPB / 32;
  __shared__ float rv[NWV];
  __shared__ int   ri[NWV];
  __shared__ int   sfar;
  const int b    = blockIdx.x;
  const int t    = threadIdx.x;
  const int wv   = t >> 5;
  const int lane = t & 31;
  float dmin[PPT];
#pragma unroll
  for (int i = 0; i < PPT; ++i) dmin[i] = 1e10f;
  int far = 0;
  for (int it = 0; it < npoint; ++it) {
    if (t == 0) out_idx[(size_t)b * npoint + it] = far;
    const float cx = xyz[((size_t)b * N + far) * 3 + 0];
    const float cy = xyz[((size_t)b * N + far) * 3 + 1];
    const float cz = xyz[((size_t)b * N + far) * 3 + 2];
    float best = -1.f;
    int bi = 0;
#pragma unroll
    for (int i = 0; i < PPT; ++i) {
      const int p = t + i * TPB;
      const float dx = xyz[((size_t)b * N + p) * 3 + 0] - cx;
      const float dy = xyz[((size_t)b * N + p) * 3 + 1] - cy;
      const float dz = xyz[((size_t)b * N + p) * 3 + 2] - cz;
      const float d = dx * dx + dy * dy + dz * dz;
      dmin[i] = fminf(dmin[i], d);
      if (dmin[i] > best) { best = dmin[i]; bi = p; }
    }
    // intra-wave argmax (no barriers)
#pragma unroll
    for (int mm = 16; mm > 0; mm >>= 1) {
      const float ov = __shfl_xor(best, mm, 32);
      const int   oj = __shfl_xor(bi, mm, 32);
      if (ov > best) { best = ov; bi = oj; }
    }
    if (lane == 0) { rv[wv] = best; ri[wv] = bi; }
    __syncthreads();
    if (wv == 0) {
      float v = (lane < NWV) ? rv[lane] : -1.f;
      int   j = (lane < NWV) ? ri[lane] : 0;
#pragma unroll
      for (int mm = 16; mm > 0; mm >>= 1) {
        const float ov = __shfl_xor(v, mm, 32);
        const int   oj = __shfl_xor(j, mm, 32);
        if (ov > v) { v = ov; j = oj; }
      }
      if (lane == 0) sfar = j;
    }
    __syncthreads();
    far = sfar;
    // no 3rd barrier needed: next write to rv/sfar is after the next barrier
  }
}

__global__ void gather_xyz_kernel(const float* __restrict__ xyz, int N,
                                  const int* __restrict__ idx, int S, int total3,
                                  float* __restrict__ out)
{
  const int g = blockIdx.x * blockDim.x + threadIdx.x;
  if (g >= total3) return;
  const int cc = g % 3;
  const int ps = g / 3;
  const int b  = ps / S;
  const int s  = ps % S;
  const int id = idx[(size_t)b * S + s];
  out[g] = xyz[((size_t)b * N + id) * 3 + cc];
}

// first nsample=32 points (index order) within radius, padded with first hit
__global__ void ball_kernel(const float* __restrict__ xyz, int N,
                            const float* __restrict__ nxyz, int S, int total,
                            float r2, int* __restrict__ out)
{
  const int g = blockIdx.x * blockDim.x + threadIdx.x;
  if (g >= total) return;
  const int b = g / S;
  const float cx = nxyz[(size_t)g * 3 + 0];
  const float cy = nxyz[(size_t)g * 3 + 1];
  const float cz = nxyz[(size_t)g * 3 + 2];
  const size_t base = (size_t)g * 32;
  int cnt = 0;
  for (int p = 0; p < N && cnt < 32; ++p) {
    const float dx = xyz[((size_t)b * N + p) * 3 + 0] - cx;
    const float dy = xyz[((size_t)b * N + p) * 3 + 1] - cy;
    const float dz = xyz[((size_t)b * N + p) * 3 + 2] - cz;
    if (dx * dx + dy * dy + dz * dz <= r2) out[base + cnt++] = p;
  }
  const int f0 = (cnt > 0) ? out[base] : 0;
  for (; cnt < 32; ++cnt) out[base + cnt] = f0;
}

// 3-NN with inverse-distance weights, normalized
__global__ void knn3_kernel(const float* __restrict__ unk, int n,
                            const float* __restrict__ kn, int m, int total,
                            int* __restrict__ oi, float* __restrict__ ow)
{
  const int g = blockIdx.x * blockDim.x + threadIdx.x;
  if (g >= total) return;
  const int b = g / n;
  const float px = unk[(size_t)g * 3 + 0];
  const float py = unk[(size_t)g * 3 + 1];
  const float pz = unk[(size_t)g * 3 + 2];
  float d0 = 3.4e38f, d1 = 3.4e38f, d2 = 3.4e38f;
  int i0 = 0, i1 = 0, i2 = 0;
  for (int j = 0; j < m; ++j) {
    const float dx = kn[((size_t)b * m + j) * 3 + 0] - px;
    const float dy = kn[((size_t)b * m + j) * 3 + 1] - py;
    const float dz = kn[((size_t)b * m + j) * 3 + 2] - pz;
    const float d = dx * dx + dy * dy + dz * dz;
    if (d < d0)      { d2 = d1; i2 = i1; d1 = d0; i1 = i0; d0 = d; i0 = j; }
    else if (d < d1) { d2 = d1; i2 = i1; d1 = d;  i1 = j; }
    else if (d < d2) { d2 = d;  i2 = j; }
  }
  const float w0 = 1.f / (d0 + 1e-8f);
  const float w1 = 1.f / (d1 + 1e-8f);
  const float w2 = 1.f / (d2 + 1e-8f);
  const float inv = 1.f / (w0 + w1 + w2);
  const size_t base = (size_t)g * 3;
  oi[base] = i0; oi[base + 1] = i1; oi[base + 2] = i2;
  ow[base] = w0 * inv; ow[base + 1] = w1 * inv; ow[base + 2] = w2 * inv;
}

// ---------------------------------------------------------------------------
// Fused SA module: gather grouped (rel-xyz ++ feats) into LDS, run the whole
// MLP chain with LDS ping-pong, fuse max-pool-over-32 into the last layer via
// atomicMax on float bits (outputs pre-zeroed, relu >= 0).
// ---------------------------------------------------------------------------
template <int ROWS, int ASTR, int BSTR>
__global__ __launch_bounds__(256) void sa_mlp_kernel(
    const float* __restrict__ xyz, int N,
    const float* __restrict__ prev, int Cprev,
    const float* __restrict__ nxyz, const int* __restrict__ bidx,
    int S, Mlp3 mp, float* __restrict__ outf)
{
  static_assert(ROWS % 16 == 0, "ROWS must be multiple of 16");
  __shared__ float XA[ROWS * ASTR];
  __shared__ float XB[ROWS * BSTR];
  const int tid = threadIdx.x;
  const size_t r0 = (size_t)blockIdx.x * ROWS;
  const int b  = (int)(r0 / ((size_t)S * 32));
  const int rb = (int)(r0 % ((size_t)S * 32));
  const int cin0 = 3 + Cprev;
  const int kp0  = (cin0 + 3) & ~3;
  for (int i = tid; i < ROWS * kp0; i += 256) {
    const int r  = i / kp0, cc = i - r * kp0;
    const int gr = rb + r;
    const int s  = gr >> 5, k = gr & 31;
    const int gi = bidx[((size_t)b * S + s) * 32 + k];
    float v = 0.f;
    if (cc < 3)
      v = xyz[((size_t)b * N + gi) * 3 + cc] - nxyz[((size_t)b * S + s) * 3 + cc];
    else if (cc - 3 < Cprev)
      v = prev[((size_t)b * N + gi) * Cprev + (cc - 3)];
    XA[r * ASTR + cc] = v;
  }
  __syncthreads();
  // explicit ping-pong pointers (NOT an initialized pointer array: that forms
  // an addrspacecast static initializer which ld.lld rejects for gfx1250)
  float* Xc = XA;  int lc = ASTR;
  float* Xn = XB;  int ln = BSTR;
  for (int L = 0; L < mp.nl; ++L) {
    if (L + 1 < mp.nl) {
      gemm_layer<OUT_LDS>(Xc, lc, Xn, ln,
                          nullptr, 0, 0, nullptr, 0, 0,
                          ROWS, mp.cin[L], mp.cout[L], mp.W[L], mp.S[L], mp.Bv[L]);
      float* tp = Xc; Xc = Xn; Xn = tp;
      int ti = lc; lc = ln; ln = ti;
    } else {
      int* gm = (int*)(outf + (size_t)b * S * mp.cout[L]);
      gemm_layer<OUT_GMAX>(Xc, lc, nullptr, 0,
                           nullptr, 0, 0, gm, mp.cout[L], rb,
                           ROWS, mp.cin[L], mp.cout[L], mp.W[L], mp.S[L], mp.Bv[L]);
    }
    __syncthreads();
  }
}

// ---------------------------------------------------------------------------
// Fused FP module: 3-NN interpolate + concat into LDS, MLP chain; HEAD variant
// additionally runs fc(128->128) as a 4th layer plus seg head + softmax.
// ---------------------------------------------------------------------------
template <int ROWS, int ASTR, int BSTR, bool HEAD>
__global__ __launch_bounds__(256) void fp_mlp_kernel(
    const int* __restrict__ kidx, const float* __restrict__ kw, int n,
    const float* __restrict__ knf, int Ckn, int m,
    const float* __restrict__ unkf, int Cunk,
    Mlp3 mp,
    const float* __restrict__ segW, const float* __restrict__ segB,
    float* __restrict__ outp)
{
  static_assert(ROWS % 16 == 0, "ROWS must be multiple of 16");
  __shared__ float XA[ROWS * ASTR];
  __shared__ float XB[ROWS * BSTR];
  const int tid = threadIdx.x;
  const int nb  = n / ROWS;
  const int b   = blockIdx.x / nb;
  const int p0  = (blockIdx.x % nb) * ROWS;
  const int cin0 = Ckn + Cunk;
  const int kp0  = (cin0 + 3) & ~3;
  for (int i = tid; i < ROWS * kp0; i += 256) {
    const int r = i / kp0, cc = i - r * kp0;
    const int p = p0 + r;
    float v = 0.f;
    if (cc < Ckn) {
      const size_t pb = ((size_t)b * n + p) * 3;
      const int j0 = kidx[pb], j1 = kidx[pb + 1], j2 = kidx[pb + 2];
      const float w0 = kw[pb], w1 = kw[pb + 1], w2 = kw[pb + 2];
      v = w0 * knf[((size_t)b * m + j0) * Ckn + cc]
        + w1 * knf[((size_t)b * m + j1) * Ckn + cc]
        + w2 * knf[((size_t)b * m + j2) * Ckn + cc];
    } else if (cc - Ckn < Cunk) {
      v = unkf[((size_t)b * n + p) * Cunk + (cc - Ckn)];
    }
    XA[r * ASTR + cc] = v;
  }
  __syncthreads();
  float* Xc = XA;  int lc = ASTR;
  float* Xn = XB;  int ln = BSTR;
  for (int L = 0; L < mp.nl; ++L) {
    const bool lastGlobal = (!HEAD) && (L + 1 == mp.nl);
    if (!lastGlobal) {
      gemm_layer<OUT_LDS>(Xc, lc, Xn, ln,
                          nullptr, 0, 0, nullptr, 0, 0,
                          ROWS, mp.cin[L], mp.cout[L], mp.W[L], mp.S[L], mp.Bv[L]);
      float* tp = Xc; Xc = Xn; Xn = tp;
      int ti = lc; lc = ln; ln = ti;
    } else {
      float* g = outp + (size_t)b * n * mp.cout[L];
      gemm_layer<OUT_GLOBAL>(Xc, lc, nullptr, 0,
                             g, mp.cout[L], p0, nullptr, 0, 0,
                             ROWS, mp.cin[L], mp.cout[L], mp.W[L], mp.S[L], mp.Bv[L]);
    }
    __syncthreads();
  }
  if (HEAD) {
    // seg head + softmax: waves take rows, lanes split the 128 channels
    const float* X = Xc;     // last layer was OUT_LDS, swapped into Xc
    const int ldx  = lc;
    const int C    = mp.cout[mp.nl - 1];
    const int wv   = tid >> 5;
    const int lane = tid & 31;
    for (int r = wv; r < ROWS; r += NW) {
      const float* xr = X + (size_t)r * ldx;
      float a0 = 0.f, a1 = 0.f;
      for (int cc = lane; cc < C; cc += 32) {
        const float x = xr[cc];
        a0 += segW[cc] * x;
        a1 += segW[C + cc] * x;
      }
#pragma unroll
      for (int mm = 16; mm > 0; mm >>= 1) {
        a0 += __shfl_xor(a0, mm, 32);
        a1 += __shfl_xor(a1, mm, 32);
      }
      if (lane == 0) {
        const float l0 = a0 + segB[0];
        const float l1 = a1 + segB[1];
        const float mx  = fmaxf(l0, l1);
        const float e0  = __expf(l0 - mx), e1 = __expf(l1 - mx);
        const float inv = 1.f / (e0 + e1);
        outp[((size_t)b * n + (p0 + r)) * 2 + 0] = e0 * inv;
        outp[((size_t)b * n + (p0 + r)) * 2 + 1] = e1 * inv;
      }
    }
  }
}

// ---------------------------------------------------------------------------
// Host launcher
// ---------------------------------------------------------------------------
extern "C" void kernel_launch(void* const* d_in, const int* in_sizes, int n_in,
                              void* d_out, int out_size, void* d_ws, size_t ws_size,
                              hipStream_t stream)
{
  (void)in_sizes; (void)n_in; (void)out_size; (void)ws_size;
  const int B = 2, N0 = 16384;
  const float* xyz = (const float*)d_in[0];
  // d_in[1] = features (0 channels), unused

  int c = 2;
  const float *saW[4][3], *saS[4][3], *saB[4][3];
  for (int l = 0; l < 4; ++l)
    for (int j = 0; j < 3; ++j) {
      saW[l][j] = (const float*)d_in[c++];
      saS[l][j] = (const float*)d_in[c++];
      saB[l][j] = (const float*)d_in[c++];
    }
  const float *fpW[4][3], *fpS[4][3], *fpB[4][3];
  const int fpnl[4] = {3, 2, 2, 2};
  for (int l = 0; l < 4; ++l)
    for (int j = 0; j < fpnl[l]; ++j) {
      fpW[l][j] = (const float*)d_in[c++];
      fpS[l][j] = (const float*)d_in[c++];
      fpB[l][j] = (const float*)d_in[c++];
    }
  const float* fcW  = (const float*)d_in[c++];
  const float* fcS  = (const float*)d_in[c++];
  const float* fcB  = (const float*)d_in[c++];
  const float* segW = (const float*)d_in[c++];
  const float* segB = (const float*)d_in[c++];

  // --- workspace layout ---
  size_t off = 0;
  auto wsa = [&](size_t bytes) -> void* {
    void* p = (char*)d_ws + off;
    off += (bytes + 255) & ~(size_t)255;
    return p;
  };
  float* l1x = (float*)wsa((size_t)B * 1024 * 3 * 4);
  float* l2x = (float*)wsa((size_t)B * 256 * 3 * 4);
  float* l3x = (float*)wsa((size_t)B * 64 * 3 * 4);
  float* l4x = (float*)wsa((size_t)B * 16 * 3 * 4);
  float* f1  = (float*)wsa((size_t)B * 1024 * 64 * 4);
  float* f2  = (float*)wsa((size_t)B * 256 * 128 * 4);
  float* f3  = (float*)wsa((size_t)B * 64 * 256 * 4);
  float* f4  = (float*)wsa((size_t)B * 16 * 512 * 4);
  float* g3  = (float*)wsa((size_t)B * 64 * 256 * 4);    // fp out @ level3
  float* g2  = (float*)wsa((size_t)B * 256 * 256 * 4);   // fp out @ level2
  float* g1  = (float*)wsa((size_t)B * 1024 * 128 * 4);  // fp out @ level1
  int*   fidx  = (int*)wsa((size_t)B * 1024 * 4);
  int*   ball1 = (int*)wsa((size_t)B * 1024 * 32 * 4);
  int*   ball2 = (int*)wsa((size_t)B * 256 * 32 * 4);
  int*   ball3 = (int*)wsa((size_t)B * 64 * 32 * 4);
  int*   ball4 = (int*)wsa((size_t)B * 16 * 32 * 4);
  int*   kidx  = (int*)wsa((size_t)B * 16384 * 3 * 4);
  float* kwv   = (float*)wsa((size_t)B * 16384 * 3 * 4);

  // --- MLP param packs ---
  const int saDims[4][4] = {{3,32,32,64},{67,64,64,128},{131,128,128,256},{259,256,256,512}};
  Mlp3 msa[4];
  for (int l = 0; l < 4; ++l) {
    msa[l].nl = 3;
    for (int j = 0; j < 3; ++j) {
      msa[l].W[j] = saW[l][j]; msa[l].S[j] = saS[l][j]; msa[l].Bv[j] = saB[l][j];
      msa[l].cin[j] = saDims[l][j]; msa[l].cout[j] = saDims[l][j + 1];
    }
  }
  const int fpDims[4][5] = {{128,128,128,128,0},{320,256,128,0,0},{384,256,256,0,0},{768,256,256,0,0}};
  Mlp3 mfp[4];
  for (int l = 0; l < 4; ++l) {
    mfp[l].nl = fpnl[l];
    for (int j = 0; j < fpnl[l]; ++j) {
      mfp[l].W[j] = fpW[l][j]; mfp[l].S[j] = fpS[l][j]; mfp[l].Bv[j] = fpB[l][j];
      mfp[l].cin[j] = fpDims[l][j]; mfp[l].cout[j] = fpDims[l][j + 1];
    }
  }
  Mlp3 mhead = mfp[0];        // [128,128,128,128] + fc(128->128) as 4th layer
  mhead.W[3] = fcW; mhead.S[3] = fcS; mhead.Bv[3] = fcB;
  mhead.cin[3] = 128; mhead.cout[3] = 128; mhead.nl = 4;

  // ================= SA1: N=16384 -> S=1024, r=0.1, mlp 3->32->32->64 ========
  fps_kernel<1024, 16><<<B, 1024, 0, stream>>>(xyz, N0, 1024, fidx);
  gather_xyz_kernel<<<(B * 1024 * 3 + 255) / 256, 256, 0, stream>>>(xyz, N0, fidx, 1024, B * 1024 * 3, l1x);
  ball_kernel<<<(B * 1024 + 255) / 256, 256, 0, stream>>>(xyz, N0, l1x, 1024, B * 1024, 0.01f, ball1);
  (void)hipMemsetAsync(f1, 0, (size_t)B * 1024 * 64 * 4, stream);
  sa_mlp_kernel<128, 32, 32><<<(B * 1024 * 32) / 128, 256, 0, stream>>>(
      xyz, N0, nullptr, 0, l1x, ball1, 1024, msa[0], f1);

  // ================= SA2: 1024 -> 256, r=0.2, mlp 67->64->64->128 ============
  fps_kernel<1024, 1><<<B, 1024, 0, stream>>>(l1x, 1024, 256, fidx);
  gather_xyz_kernel<<<(B * 256 * 3 + 255) / 256, 256, 0, stream>>>(l1x, 1024, fidx, 256, B * 256 * 3, l2x);
  ball_kernel<<<(B * 256 + 255) / 256, 256, 0, stream>>>(l1x, 1024, l2x, 256, B * 256, 0.04f, ball2);
  (void)hipMemsetAsync(f2, 0, (size_t)B * 256 * 128 * 4, stream);
  sa_mlp_kernel<64, 68, 64><<<(B * 256 * 32) / 64, 256, 0, stream>>>(
      l1x, 1024, f1, 64, l2x, ball2, 256, msa[1], f2);

  // ================= SA3: 256 -> 64, r=0.4, mlp 131->128->128->256 ===========
  fps_kernel<256, 1><<<B, 256, 0, stream>>>(l2x, 256, 64, fidx);
  gather_xyz_kernel<<<(B * 64 * 3 + 255) / 256, 256, 0, stream>>>(l2x, 256, fidx, 64, B * 64 * 3, l3x);
  ball_kernel<<<(B * 64 + 255) / 256, 256, 0, stream>>>(l2x, 256, l3x, 64, B * 64, 0.16f, ball3);
  (void)hipMemsetAsync(f3, 0, (size_t)B * 64 * 256 * 4, stream);
  sa_mlp_kernel<32, 132, 128><<<(B * 64 * 32) / 32, 256, 0, stream>>>(
      l2x, 256, f2, 128, l3x, ball3, 64, msa[2], f3);

  // ================= SA4: 64 -> 16, r=0.8, mlp 259->256->256->512 ============
  fps_kernel<64, 1><<<B, 64, 0, stream>>>(l3x, 64, 16, fidx);
  gather_xyz_kernel<<<(B * 16 * 3 + 255) / 256, 256, 0, stream>>>(l3x, 64, fidx, 16, B * 16 * 3, l4x);
  ball_kernel<<<(B * 16 + 255) / 256, 256, 0, stream>>>(l3x, 64, l4x, 16, B * 16, 0.64f, ball4);
  (void)hipMemsetAsync(f4, 0, (size_t)B * 16 * 512 * 4, stream);
  sa_mlp_kernel<16, 260, 256><<<(B * 16 * 32) / 16, 256, 0, stream>>>(
      l3x, 64, f3, 256, l4x, ball4, 16, msa[3], f4);

  // ================= FP @ level3: interp(f4,512) ++ f3(256) -> [768,256,256] =
  knn3_kernel<<<(B * 64 + 255) / 256, 256, 0, stream>>>(l3x, 64, l4x, 16, B * 64, kidx, kwv);
  fp_mlp_kernel<16, 768, 256, false><<<B * (64 / 16), 256, 0, stream>>>(
      kidx, kwv, 64, f4, 512, 16, f3, 256, mfp[3], nullptr, nullptr, g3);

  // ================= FP @ level2: interp(g3,256) ++ f2(128) -> [384,256,256] =
  knn3_kernel<<<(B * 256 + 255) / 256, 256, 0, stream>>>(l2x, 256, l3x, 64, B * 256, kidx, kwv);
  fp_mlp_kernel<16, 384, 256, false><<<B * (256 / 16), 256, 0, stream>>>(
      kidx, kwv, 256, g3, 256, 64, f2, 128, mfp[2], nullptr, nullptr, g2);

  // ================= FP @ level1: interp(g2,256) ++ f1(64) -> [320,256,128] ==
  knn3_kernel<<<(B * 1024 + 255) / 256, 256, 0, stream>>>(l1x, 1024, l2x, 256, B * 1024, kidx, kwv);
  fp_mlp_kernel<16, 320, 256, false><<<B * (1024 / 16), 256, 0, stream>>>(
      kidx, kwv, 1024, g2, 256, 256, f1, 64, mfp[1], nullptr, nullptr, g1);

  // ===== FP @ level0 + head: interp(g1,128) -> [128,128,128,128]+fc+seg+softmax
  knn3_kernel<<<(B * 16384 + 255) / 256, 256, 0, stream>>>(xyz, 16384, l1x, 1024, B * 16384, kidx, kwv);
  fp_mlp_kernel<32, 128, 128, true><<<B * (16384 / 32), 256, 0, stream>>>(
      kidx, kwv, 16384, g1, 128, 1024, nullptr, 0, mhead, segW, segB, (float*)d_out);
}